// BatchedGNNLayer_2851858284626
// MI455X (gfx1250) — compile-verified
//
#include <hip/hip_runtime.h>

#define B_    8
#define N_    50000
#define E_    800000
#define DIM   64

typedef __attribute__((ext_vector_type(2))) float v2f;
typedef __attribute__((ext_vector_type(8))) float v8f;

// ---------------------------------------------------------------------------
// Kernel 1: scatter-add  agg[b, dst[e], :] += x[b, src[e], :]
// One thread per (edge, batch, 16-byte chunk): coalesced float4 loads,
// native global_atomic_add_f32 into the L2-resident agg workspace.
// ---------------------------------------------------------------------------
__global__ void __launch_bounds__(256)
gnn_scatter(const float* __restrict__ x,
            const int*   __restrict__ ei,   // (E,2) int32: [src, dst]
            float*       __restrict__ agg) {
  const long long t = (long long)blockIdx.x * 256 + threadIdx.x;
  const int       q = (int)(t & 15);        // which float4 of the 64-dim row
  const int       b = (int)((t >> 4) & 7);  // batch
  const long long e = t >> 7;               // edge
  if (e >= E_) return;

  const int s = ei[2 * e + 0];
  const int d = ei[2 * e + 1];

  const float4 v = *(const float4*)(x + ((long long)b * N_ + s) * DIM + q * 4);
  float* p = agg + ((long long)b * N_ + d) * DIM + q * 4;

  __hip_atomic_fetch_add(p + 0, v.x, __ATOMIC_RELAXED, __HIP_MEMORY_SCOPE_AGENT);
  __hip_atomic_fetch_add(p + 1, v.y, __ATOMIC_RELAXED, __HIP_MEMORY_SCOPE_AGENT);
  __hip_atomic_fetch_add(p + 2, v.z, __ATOMIC_RELAXED, __HIP_MEMORY_SCOPE_AGENT);
  __hip_atomic_fetch_add(p + 3, v.w, __ATOMIC_RELAXED, __HIP_MEMORY_SCOPE_AGENT);
}

// ---------------------------------------------------------------------------
// Kernel 2: fused  out = relu(x @ Ws^T + bs + agg @ Wn^T + bn)
// fp32 WMMA (V_WMMA_F32_16X16X4_F32). Each wave: 16 rows x 64 cols.
// B-fragments of both weight matrices pre-swizzled into LDS, fragment-major,
// so ds_load_b64 per lane is bank-conflict-free.
// ---------------------------------------------------------------------------
__global__ void __launch_bounds__(256)
gnn_gemm(const float* __restrict__ x,
         const float* __restrict__ agg,
         const float* __restrict__ Wself,
         const float* __restrict__ bself,
         const float* __restrict__ Wneigh,
         const float* __restrict__ bneigh,
         float*       __restrict__ out) {
  // frag index f = kt*4 + nt   (kt: K-tile 0..15, nt: N-tile 0..3)
  // lane l holds B[k][n], B[k+1][n] with k = kt*4 + 2*(l>>4), n = nt*16 + (l&15)
  __shared__ float2 fragS[64][32];
  __shared__ float2 fragN[64][32];

  const int tid = threadIdx.x;
  for (int i = tid; i < 64 * 32; i += 256) {
    const int f  = i >> 5;
    const int l  = i & 31;
    const int kt = f >> 2;
    const int nt = f & 3;
    const int n  = nt * 16 + (l & 15);
    const int k  = kt * 4 + 2 * (l >> 4);
    // B[k][n] = W[n][k]  (h = x @ W^T)
    fragS[f][l] = make_float2(Wself [n * DIM + k], Wself [n * DIM + k + 1]);
    fragN[f][l] = make_float2(Wneigh[n * DIM + k], Wneigh[n * DIM + k + 1]);
  }
  __syncthreads();

  const int wave = tid >> 5;
  const int lane = tid & 31;
  const int lo   = lane & 15;
  const int hi   = lane >> 4;

  const long long rowbase = ((long long)blockIdx.x * 8 + wave) * 16;
  // A-fragment base: row (rowbase+lo), k starts at 2*hi (ISA 16x4 fp32 layout)
  const long long abase = (rowbase + lo) * DIM + 2 * hi;

  v8f acc[4] = {v8f{}, v8f{}, v8f{}, v8f{}};

#pragma unroll
  for (int kt = 0; kt < 16; ++kt) {
    const long long off = abase + 4 * kt;
    const v2f ax = *(const v2f*)(x + off);
    const v2f ag = *(const v2f*)(agg + off);
    const int fb = kt * 4;
#pragma unroll
    for (int nt = 0; nt < 4; ++nt) {
      const v2f bS = *(const v2f*)&fragS[fb + nt][lane];
      const v2f bN = *(const v2f*)&fragN[fb + nt][lane];
      acc[nt] = __builtin_amdgcn_wmma_f32_16x16x4_f32(
          false, ax, false, bS, (short)0, acc[nt], false, false);
      acc[nt] = __builtin_amdgcn_wmma_f32_16x16x4_f32(
          false, ag, false, bN, (short)0, acc[nt], false, false);
    }
  }

  // Epilogue: bias + relu + non-temporal store.
  // C/D layout: VGPR r -> M = r (lanes 0-15) / M = r+8 (lanes 16-31), N = lane&15
#pragma unroll
  for (int nt = 0; nt < 4; ++nt) {
    const int   col  = nt * 16 + lo;
    const float bias = bself[col] + bneigh[col];
#pragma unroll
    for (int r = 0; r < 8; ++r) {
      const long long row = rowbase + r + 8 * hi;
      const float v = fmaxf(acc[nt][r] + bias, 0.0f);
      __builtin_nontemporal_store(v, out + row * DIM + col);
    }
  }
}

// ---------------------------------------------------------------------------
extern "C" void kernel_launch(void* const* d_in, const int* in_sizes, int n_in,
                              void* d_out, int out_size, void* d_ws, size_t ws_size,
                              hipStream_t stream) {
  const float* x  = (const float*)d_in[0];
  const int*   ei = (const int*)  d_in[1];   // (E,2) int32
  const float* Ws = (const float*)d_in[2];
  const float* bs = (const float*)d_in[3];
  const float* Wn = (const float*)d_in[4];
  const float* bn = (const float*)d_in[5];
  float*       out = (float*)d_out;
  float*       agg = (float*)d_ws;           // B*N*DIM floats = 102.4 MB

  const size_t aggBytes = (size_t)B_ * N_ * DIM * sizeof(float);
  hipMemsetAsync(agg, 0, aggBytes, stream);

  const long long tscatter = (long long)E_ * B_ * 16;            // 102.4M threads
  gnn_scatter<<<(unsigned)((tscatter + 255) / 256), 256, 0, stream>>>(x, ei, agg);

  const long long rows = (long long)B_ * N_;                     // 400000
  gnn_gemm<<<(unsigned)(rows / 128), 256, 0, stream>>>(x, agg, Ws, bs, Wn, bn, out);
}